// Original_model_39968965657064
// MI455X (gfx1250) — compile-verified
//
#include <hip/hip_runtime.h>
#include <hip/hip_bf16.h>
#include <math.h>

typedef __attribute__((ext_vector_type(2))) float v2f;
typedef __attribute__((ext_vector_type(8))) float v8f;

#define NNODES 100000
#define NEDGES 3200000
#define NFEAT  128
#define NGRAPH 512
#define HID    64

// ---------------- utility ----------------
__global__ void fill_zero(float* __restrict__ p, long long n) {
  long long i = (long long)blockIdx.x * blockDim.x + threadIdx.x;
  if (i < n) p[i] = 0.0f;
}

// ---------------- degree / norm ----------------
__global__ void degree_kernel(const long long* __restrict__ dst,
                              float* __restrict__ deg, int nE) {
  int e = blockIdx.x * blockDim.x + threadIdx.x;
  if (e < nE) atomicAdd(&deg[dst[e]], 1.0f);
}

__global__ void dinv_kernel(float* __restrict__ deg, int n) {
  int i = blockIdx.x * blockDim.x + threadIdx.x;
  if (i < n) deg[i] = rsqrtf(deg[i] + 1.0f);  // +1 = self loop; always > 0
}

// ---------------- fp32 WMMA GEMM:  C[M,64] = A[M,K] @ B[K,64] ----------------
// One wave computes a full 16-row x 64-col slab: 4 independent 16x16
// accumulator tiles -> 4 independent V_WMMA_F32_16X16X4_F32 chains per K-step
// (breaks the serial acc RAW chain, lets XDL issue back-to-back), and the A
// fragment is loaded once per wave instead of once per N-tile.
// Fragment layouts per ISA 7.12.2 (32-bit tables).
__global__ __launch_bounds__(128) void gemm_wmma_f32(
    const float* __restrict__ A, const float* __restrict__ B,
    float* __restrict__ C, int M, int K) {
  const int lane    = threadIdx.x & 31;
  const int wave    = threadIdx.x >> 5;
  const int row0    = (blockIdx.x * 4 + wave) * 16;  // 16-row slab per wave
  if (row0 >= M) return;
  const int halfSel = lane >> 4;          // 0: K pair (k,k+1)  1: (k+2,k+3)
  const int l15     = lane & 15;

  v8f acc0 = {}, acc1 = {}, acc2 = {}, acc3 = {};
  const float* Arow = A + (size_t)(row0 + l15) * K;

#pragma unroll 4
  for (int k = 0; k < K; k += 4) {
    const int ka = k + 2 * halfSel;
    // stream-prefetch A ahead of consumption (global_prefetch_b8; speculative)
    __builtin_prefetch(Arow + ka + 64, 0, 1);
    v2f a;
    a.x = Arow[ka];
    a.y = Arow[ka + 1];
    const float* Brow0 = B + (size_t)ka * HID + l15;
    const float* Brow1 = B + (size_t)(ka + 1) * HID + l15;
    v2f b0, b1, b2, b3;
    b0.x = Brow0[0];  b0.y = Brow1[0];
    b1.x = Brow0[16]; b1.y = Brow1[16];
    b2.x = Brow0[32]; b2.y = Brow1[32];
    b3.x = Brow0[48]; b3.y = Brow1[48];
    acc0 = __builtin_amdgcn_wmma_f32_16x16x4_f32(false, a, false, b0, (short)0, acc0, false, false);
    acc1 = __builtin_amdgcn_wmma_f32_16x16x4_f32(false, a, false, b1, (short)0, acc1, false, false);
    acc2 = __builtin_amdgcn_wmma_f32_16x16x4_f32(false, a, false, b2, (short)0, acc2, false, false);
    acc3 = __builtin_amdgcn_wmma_f32_16x16x4_f32(false, a, false, b3, (short)0, acc3, false, false);
  }

  // D layout: VGPR r -> M = r + 8*halfSel, N = n0 + l15
  float* Crow = C + (size_t)(row0 + halfSel * 8) * HID + l15;
#pragma unroll
  for (int r = 0; r < 8; ++r) {
    Crow[(size_t)r * HID +  0] = acc0[r];
    Crow[(size_t)r * HID + 16] = acc1[r];
    Crow[(size_t)r * HID + 32] = acc2[r];
    Crow[(size_t)r * HID + 48] = acc3[r];
  }
}

// ---------------- edge scatter-add: out[dst] += h[src] * dinv[s]*dinv[d] ----
// 16 threads per edge, float4 (16B) per thread -> 64 feats. h/out tables are
// L2-resident (25.6 MB each vs 192 MB L2); atomics resolve in L2.
__global__ void edge_agg(const long long* __restrict__ src,
                         const long long* __restrict__ dst,
                         const float* __restrict__ h,
                         const float* __restrict__ dinv,
                         float* __restrict__ out, int nE) {
  long long gid = (long long)blockIdx.x * blockDim.x + threadIdx.x;
  long long e = gid >> 4;
  if (e >= nE) return;
  int fo = (int)(gid & 15) * 4;
  long long s = src[e], d = dst[e];
  float norm = dinv[s] * dinv[d];
  float4 v = *(const float4*)(h + s * HID + fo);
  float* o = out + d * HID + fo;
  atomicAdd(o + 0, v.x * norm);
  atomicAdd(o + 1, v.y * norm);
  atomicAdd(o + 2, v.z * norm);
  atomicAdd(o + 3, v.w * norm);
}

// ---------------- finalize: agg += h*dinv^2 (self loop) + bias; opt relu ----
__global__ void finalize_kernel(float* __restrict__ agg,
                                const float* __restrict__ h,
                                const float* __restrict__ dinv,
                                const float* __restrict__ bias,
                                int relu, long long n) {
  long long gid = (long long)blockIdx.x * blockDim.x + threadIdx.x;
  if (gid >= n) return;
  int node = (int)(gid >> 6);
  int f = (int)(gid & 63);
  float di = dinv[node];
  float v = agg[gid] + h[gid] * di * di + bias[f];
  if (relu) v = fmaxf(v, 0.0f);
  agg[gid] = v;
}

// ---------------- mean pool over graphs ----------------
__global__ void pool_kernel(const float* __restrict__ h,
                            const long long* __restrict__ batch,
                            float* __restrict__ pooled,
                            float* __restrict__ counts, long long n) {
  long long gid = (long long)blockIdx.x * blockDim.x + threadIdx.x;
  if (gid >= n) return;
  int node = (int)(gid >> 6);
  int f = (int)(gid & 63);
  long long b = batch[node];
  atomicAdd(&pooled[b * HID + f], h[gid]);
  if (f == 0) atomicAdd(&counts[b], 1.0f);
}

// ---------------- MLP head + softmax over dim 0 ----------------
__global__ __launch_bounds__(NGRAPH) void head_kernel(
    const float* __restrict__ pooled, const float* __restrict__ counts,
    const float* __restrict__ Wl1, const float* __restrict__ bl1,
    const float* __restrict__ Wl2, const float* __restrict__ bl2,
    float* __restrict__ out) {
  __shared__ float sZ[NGRAPH * 2];
  __shared__ float sMS[4];  // m0, m1, s0, s1
  int g = threadIdx.x;

  float inv = 1.0f / fmaxf(counts[g], 1.0f);
  float p[HID];
#pragma unroll
  for (int f = 0; f < HID; ++f) p[f] = pooled[g * HID + f] * inv;

  float z0 = bl2[0], z1 = bl2[1];
  for (int j = 0; j < 32; ++j) {
    float a = bl1[j];
#pragma unroll
    for (int f = 0; f < HID; ++f) a += p[f] * Wl1[f * 32 + j];
    z0 += a * Wl2[j * 2 + 0];
    z1 += a * Wl2[j * 2 + 1];
  }
  sZ[g * 2 + 0] = z0;
  sZ[g * 2 + 1] = z1;
  __syncthreads();

  if (g < 2) {
    float m = -INFINITY;
    for (int i = 0; i < NGRAPH; ++i) m = fmaxf(m, sZ[i * 2 + g]);
    float s = 0.0f;
    for (int i = 0; i < NGRAPH; ++i) s += expf(sZ[i * 2 + g] - m);
    sMS[g] = m;
    sMS[2 + g] = s;
  }
  __syncthreads();

  out[g * 2 + 0] = expf(sZ[g * 2 + 0] - sMS[0]) / sMS[2];
  out[g * 2 + 1] = expf(sZ[g * 2 + 1] - sMS[1]) / sMS[3];
}

// ---------------- driver ----------------
extern "C" void kernel_launch(void* const* d_in, const int* in_sizes, int n_in,
                              void* d_out, int out_size, void* d_ws, size_t ws_size,
                              hipStream_t stream) {
  const float*     x     = (const float*)d_in[0];
  const long long* ei    = (const long long*)d_in[1];   // int64 [2, E]
  const long long* src   = ei;
  const long long* dst   = ei + NEDGES;
  const long long* batch = (const long long*)d_in[2];   // int64 [N]
  const float* W1  = (const float*)d_in[3];
  const float* b1  = (const float*)d_in[4];
  const float* W2  = (const float*)d_in[5];
  const float* b2  = (const float*)d_in[6];
  const float* Wl1 = (const float*)d_in[7];
  const float* bl1 = (const float*)d_in[8];
  const float* Wl2 = (const float*)d_in[9];
  const float* bl2 = (const float*)d_in[10];
  float* out = (float*)d_out;

  char* ws = (char*)d_ws;
  size_t off = 0;
  auto carve = [&](size_t bytes) -> float* {
    float* p = (float*)(ws + off);
    off = (off + bytes + 255) & ~(size_t)255;
    return p;
  };
  float* dinv   = carve((size_t)NNODES * 4);          // deg -> dinv in place
  float* bufH   = carve((size_t)NNODES * HID * 4);    // GEMM outputs
  float* bufA   = carve((size_t)NNODES * HID * 4);    // aggregation accumulator
  float* pooled = carve((size_t)NGRAPH * HID * 4);
  float* counts = carve((size_t)NGRAPH * 4);

  const long long nElems = (long long)NNODES * HID;
  const int bElems = (int)((nElems + 255) / 256);
  const long long eThreads = (long long)NEDGES * 16;
  const int bEdges = (int)((eThreads + 255) / 256);
  const int nSlabs = NNODES / 16;                     // 6250
  const int bGemm  = (nSlabs + 3) / 4;                // 4 waves/block

  // normalization: deg (in-degree) -> dinv = rsqrt(deg+1)
  fill_zero<<<(NNODES + 255) / 256, 256, 0, stream>>>(dinv, NNODES);
  degree_kernel<<<(NEDGES + 255) / 256, 256, 0, stream>>>(dst, dinv, NEDGES);
  dinv_kernel<<<(NNODES + 255) / 256, 256, 0, stream>>>(dinv, NNODES);

  // layer 1: h = x @ W1 ; agg ; +self loop +bias ; relu
  gemm_wmma_f32<<<bGemm, 128, 0, stream>>>(x, W1, bufH, NNODES, NFEAT);
  fill_zero<<<bElems, 256, 0, stream>>>(bufA, nElems);
  edge_agg<<<bEdges, 256, 0, stream>>>(src, dst, bufH, dinv, bufA, NEDGES);
  finalize_kernel<<<bElems, 256, 0, stream>>>(bufA, bufH, dinv, b1, 1, nElems);

  // layer 2: h = relu_out @ W2 ; agg ; +self loop +bias (no relu)
  gemm_wmma_f32<<<bGemm, 128, 0, stream>>>(bufA, W2, bufH, NNODES, HID);
  fill_zero<<<bElems, 256, 0, stream>>>(bufA, nElems);
  edge_agg<<<bEdges, 256, 0, stream>>>(src, dst, bufH, dinv, bufA, NEDGES);
  finalize_kernel<<<bElems, 256, 0, stream>>>(bufA, bufH, dinv, b2, 0, nElems);

  // global mean pool + MLP head + softmax(dim 0)
  fill_zero<<<((NGRAPH * HID) + 255) / 256, 256, 0, stream>>>(pooled, NGRAPH * HID);
  fill_zero<<<(NGRAPH + 255) / 256, 256, 0, stream>>>(counts, NGRAPH);
  pool_kernel<<<bElems, 256, 0, stream>>>(bufA, batch, pooled, counts, nElems);
  head_kernel<<<1, NGRAPH, 0, stream>>>(pooled, counts, Wl1, bl1, Wl2, bl2, out);
}